// skipgram_41145786695829
// MI455X (gfx1250) — compile-verified
//
#include <hip/hip_runtime.h>

typedef __attribute__((ext_vector_type(2))) float v2f;
typedef __attribute__((ext_vector_type(8))) float v8f;

#define SG_DIM 128
#define SG_LSTRIDE 132   // 128 + 4 dword pad -> bank-conflict-free column reads
#define SG_KNEG 20

__device__ __forceinline__ float log_sigmoid_f(float x) {
    // numerically stable: min(x,0) - log1p(exp(-|x|))
    return fminf(x, 0.0f) - log1pf(__expf(-fabsf(x)));
}

// One wave32 per tile of 16 batch elements.
__global__ __launch_bounds__(32)
void sg_tile_kernel(const float* __restrict__ u_emb,
                    const float* __restrict__ v_emb,
                    const int*   __restrict__ u_pos,
                    const int*   __restrict__ v_pos,
                    const int*   __restrict__ v_neg,
                    float* __restrict__ ws_partial,
                    int K)
{
    __shared__ __align__(16) float sU[16 * SG_LSTRIDE];
    __shared__ __align__(16) float sV[16 * SG_LSTRIDE];
    __shared__ __align__(16) float sW[16 * SG_LSTRIDE];
    __shared__ int sUp[16];
    __shared__ int sVp[16];
    __shared__ int sNeg[16 * SG_KNEG];

    const int lane = threadIdx.x;          // 0..31
    const int tile = blockIdx.x;
    const int b0   = tile * 16;

    // ---- stage indices into LDS ----
    if (lane < 16) {
        sUp[lane] = u_pos[b0 + lane];
        sVp[lane] = v_pos[b0 + lane];
    }
    for (int t = lane; t < 16 * K; t += 32)
        sNeg[t] = v_neg[b0 * K + t];       // 16 rows of K are contiguous
    __syncthreads();

    // ---- gather rows: one global_load_b128 per lane covers one 512B row ----
    for (int m = 0; m < 16; ++m) {
        const float4 uu = ((const float4*)(u_emb + (long)sUp[m] * SG_DIM))[lane];
        ((float4*)(&sU[m * SG_LSTRIDE]))[lane] = uu;

        const float4 vv = ((const float4*)(v_emb + (long)sVp[m] * SG_DIM))[lane];
        ((float4*)(&sV[m * SG_LSTRIDE]))[lane] = vv;

        float4 acc; acc.x = 0.f; acc.y = 0.f; acc.z = 0.f; acc.w = 0.f;
        for (int k = 0; k < K; ++k) {
            const float4 nv = ((const float4*)(v_emb + (long)sNeg[m * K + k] * SG_DIM))[lane];
            acc.x += nv.x; acc.y += nv.y; acc.z += nv.z; acc.w += nv.w;
        }
        ((float4*)(&sW[m * SG_LSTRIDE]))[lane] = acc;
    }
    __syncthreads();

    // ---- WMMA accumulation: C1 = U*V^T, C2 = U*W^T over K chunks of 4 ----
    v8f c1 = {};
    v8f c2 = {};
    const int rowA   = lane & 15;
    const int colOff = 2 * (lane >> 4);    // lanes 16-31 hold K+2/K+3 per f32 A layout
#pragma unroll
    for (int kk = 0; kk < SG_DIM; kk += 4) {
        const int col = kk + colOff;
        const v2f a  = *(const v2f*)&sU[rowA * SG_LSTRIDE + col];
        const v2f bv = *(const v2f*)&sV[rowA * SG_LSTRIDE + col];
        const v2f bw = *(const v2f*)&sW[rowA * SG_LSTRIDE + col];
        c1 = __builtin_amdgcn_wmma_f32_16x16x4_f32(false, a, false, bv, (short)0, c1, false, false);
        c2 = __builtin_amdgcn_wmma_f32_16x16x4_f32(false, a, false, bw, (short)0, c2, false, false);
    }

    // ---- diagonal extraction ----
    // C/D layout: vgpr r, lanes 0-15 -> (M=r, N=lane); lanes 16-31 -> (M=8+r, N=lane-16)
    // diag: lanes 0-7 hold score[lane] in vgpr=lane; lanes 24-31 hold score[lane-16] in vgpr=lane-24
    const bool valid = (lane < 8) || (lane >= 24);
    const int  r     = (lane < 16) ? lane : (lane - 24);
    float s1 = 0.f, s2 = 0.f;
#pragma unroll
    for (int j = 0; j < 8; ++j) {
        if (r == j) { s1 = c1[j]; s2 = c2[j]; }
    }

    float contrib = 0.f;
    if (valid)
        contrib = log_sigmoid_f(s1) + log_sigmoid_f(-s2);

    // wave32 reduction
    for (int off = 16; off; off >>= 1)
        contrib += __shfl_down(contrib, off, 32);

    if (lane == 0)
        ws_partial[tile] = contrib;
}

__global__ __launch_bounds__(256)
void sg_reduce_kernel(const float* __restrict__ ws_partial, int n,
                      float neg_inv_b, float* __restrict__ out)
{
    __shared__ float sh[8];
    float s = 0.f;
    for (int i = threadIdx.x; i < n; i += 256)
        s += ws_partial[i];
    for (int off = 16; off; off >>= 1)
        s += __shfl_down(s, off, 32);
    if ((threadIdx.x & 31) == 0)
        sh[threadIdx.x >> 5] = s;
    __syncthreads();
    if (threadIdx.x == 0) {
        float t = 0.f;
        for (int i = 0; i < 8; ++i) t += sh[i];
        out[0] = t * neg_inv_b;
    }
}

extern "C" void kernel_launch(void* const* d_in, const int* in_sizes, int n_in,
                              void* d_out, int out_size, void* d_ws, size_t ws_size,
                              hipStream_t stream)
{
    const float* u_emb = (const float*)d_in[0];
    const float* v_emb = (const float*)d_in[1];
    const int*   u_pos = (const int*)d_in[2];
    const int*   v_pos = (const int*)d_in[3];
    const int*   v_neg = (const int*)d_in[4];

    const int B = in_sizes[2];             // 16384
    const int K = in_sizes[4] / B;         // 20
    const int tiles = B / 16;              // 1024

    float* ws = (float*)d_ws;

    sg_tile_kernel<<<tiles, 32, 0, stream>>>(u_emb, v_emb, u_pos, v_pos, v_neg, ws, K);
    sg_reduce_kernel<<<1, 256, 0, stream>>>(ws, tiles, -1.0f / (float)B, (float*)d_out);
}